// MultiBoxLoss_1949915153019
// MI455X (gfx1250) — compile-verified
//
#include <hip/hip_runtime.h>
#include <math.h>

typedef __attribute__((ext_vector_type(2))) float v2f;
typedef __attribute__((ext_vector_type(8))) float v8f;

#define TPB 512
#define MAXT 128
#define BIG 1e18

__device__ __forceinline__ float box_iou(const float4 a, const float4 b) {
  float area1 = (a.z - a.x) * (a.w - a.y);
  float area2 = (b.z - b.x) * (b.w - b.y);
  float ltx = fmaxf(a.x, b.x), lty = fmaxf(a.y, b.y);
  float rbx = fminf(a.z, b.z), rby = fminf(a.w, b.w);
  float w = fmaxf(rbx - ltx, 0.f), h = fmaxf(rby - lty, 0.f);
  float inter = w * h;
  return inter / (area1 + area2 - inter);
}

// Single-workgroup cooperative kernel:
//   phase 1: Hungarian (JV shortest augmenting path) over 128 rows x 100000 cols,
//            IoU recomputed on the fly (L2-resident), f64 potentials like reference.
//   phase 2: smooth-L1 + cross-entropy + accuracy over matched pairs.
//            Softmax row-sums (128x81) computed with v_wmma_f32_16x16x4_f32
//            (A = exp(logits) tile, B = ones) -- 8 waves, 21 K-chunks each.
__global__ __launch_bounds__(TPB)
void multibox_hungarian_loss(const float4* __restrict__ boxes,
                             const float*  __restrict__ confs,
                             const float4* __restrict__ tlocs,
                             const int*    __restrict__ tlabels,
                             float*        __restrict__ out,
                             double*       __restrict__ v,     // [N+1]
                             double*       __restrict__ s,     // [N+1] minv surrogate
                             int*          __restrict__ way,   // [N+1]
                             int*          __restrict__ p,     // [N+1]
                             unsigned char* __restrict__ used, // [N+1]
                             int N, int C, int Mt)
{
  const int tid = threadIdx.x;

  __shared__ float4 s_tbox[MAXT];
  __shared__ int    s_tlab[MAXT];
  __shared__ double s_u[MAXT + 1];
  __shared__ double rv[TPB];
  __shared__ int    ri[TPB];
  __shared__ int    s_usedList[MAXT + 2];
  __shared__ int    s_cnt, s_j0, s_i0, s_j1, s_done;
  __shared__ double s_H, s_delta;
  __shared__ int    s_matchPred[MAXT];
  __shared__ float  s_sumexp[MAXT];

  // ---- init ----
  for (int t = tid; t < Mt; t += TPB) {
    s_tbox[t] = tlocs[t];
    s_tlab[t] = tlabels[t];
  }
  for (int t = tid; t <= Mt; t += TPB) s_u[t] = 0.0;
  for (int j = tid; j <= N; j += TPB) { v[j] = 0.0; p[j] = 0; }
  __syncthreads();

  // ---- Hungarian: rows are targets (1..Mt), cols are predictions (1..N) ----
  for (int i = 1; i <= Mt; ++i) {
    for (int j = tid; j <= N; j += TPB) { s[j] = BIG; used[j] = 0; }
    if (tid == 0) { p[0] = i; s_j0 = 0; s_cnt = 0; s_H = 0.0; }
    __syncthreads();

    while (true) {
      if (tid == 0) {
        used[s_j0] = 1;
        s_usedList[s_cnt++] = s_j0;
        s_i0 = p[s_j0];
      }
      __syncthreads();

      const int    i0 = s_i0;
      const double ui = s_u[i0];
      const double Hh = s_H;
      const int    j0 = s_j0;
      const float4 tb = s_tbox[i0 - 1];

      // one fused scan: relax minv with current row, and argmin over free cols.
      // true_minv[j] = s[j] - H; store s[j] = cur + H_at_set so argmin is invariant.
      double lval = BIG; int lidx = 0;
      for (int j = 1 + tid; j <= N; j += TPB) {
        int jp = j - 1 + TPB; jp = (jp < N) ? jp : (N - 1);
        __builtin_prefetch(&boxes[jp], 0, 1);           // global_prefetch_b8, branchless
        if (!used[j]) {
          float  iou = box_iou(boxes[j - 1], tb);
          double cur = -(double)iou - ui - v[j] + Hh;
          double sj  = s[j];
          if (cur < sj) { s[j] = cur; way[j] = j0; sj = cur; }
          if (sj < lval) { lval = sj; lidx = j; }
        }
      }
      rv[tid] = lval; ri[tid] = lidx;
      __syncthreads();
      for (int off = TPB / 2; off > 0; off >>= 1) {
        if (tid < off) {
          double v2 = rv[tid + off]; int i2 = ri[tid + off];
          if (v2 < rv[tid] || (v2 == rv[tid] && i2 < ri[tid])) { rv[tid] = v2; ri[tid] = i2; }
        }
        __syncthreads();
      }
      if (tid == 0) {
        double mval = rv[0]; int j1 = ri[0];
        s_delta = mval - s_H;   // delta = true minv at j1
        s_H     = mval;         // H += delta
        s_j1    = j1;
        s_done  = (p[j1] == 0);
      }
      __syncthreads();
      // u[p[used]] += delta; v[used] -= delta  (compact list, <=129 entries)
      if (tid < s_cnt) {
        int jc = s_usedList[tid];
        s_u[p[jc]] += s_delta;
        v[jc]      -= s_delta;
      }
      if (tid == 0) s_j0 = s_j1;
      __syncthreads();
      if (s_done) break;
    }

    // augment along alternating path
    if (tid == 0) {
      int j0 = s_j1;
      while (j0) { int j1 = way[j0]; p[j0] = p[j1]; j0 = j1; }
    }
    __syncthreads();
  }

  // ---- extract matches: target t -> prediction s_matchPred[t] ----
  for (int j = 1 + tid; j <= N; j += TPB) {
    int r = p[j];
    if (r > 0) s_matchPred[r - 1] = j - 1;
  }
  __syncthreads();

  // ---- softmax denominators via WMMA f32 16x16x4 (rowsum of exp(logits)) ----
  {
    const int wave = tid >> 5, lane = tid & 31;
    const int nTiles = (Mt + 15) >> 4;
    if (wave < nTiles) {                       // wave-uniform: EXEC stays all-ones
      const int  m     = lane & 15;
      const int  row   = wave * 16 + m;
      const bool valid = row < Mt;
      const int  prow  = s_matchPred[valid ? row : 0];   // clamped gather, branchless
      const float* lrow = confs + (size_t)prow * (size_t)C;
      const int kbase = (lane >= 16) ? 2 : 0;  // A 16x4 f32 layout: lo lanes K0,K1; hi lanes K2,K3
      const float mA  = valid ? 1.f : 0.f;     // loop-invariant validity mask (no branches in loop)
      const v2f b = {1.f, 1.f};                // B = ones -> D[m][n] = rowsum(A[m][:])
      v8f c = {};

      // full chunks: unconditional loads + exp, masked by multiply -> EXEC untouched
      const int nFull = C >> 2;                // 20 for C=81
      for (int ch = 0; ch < nFull; ++ch) {
        int k = 4 * ch + kbase;
        float e0 = expf(lrow[k]);
        float e1 = expf(lrow[k + 1]);
        v2f a;
        a.x = e0 * mA;
        a.y = e1 * mA;
        c = __builtin_amdgcn_wmma_f32_16x16x4_f32(
                false, a, false, b, (short)0, c, false, false);
      }
      // tail chunk (K = C & ~3 .. C-1): clamped loads, multiplicative masks, no branches
      if (C & 3) {
        int k0 = (nFull << 2) + kbase;
        int k1 = k0 + 1;
        float m0 = (k0 < C) ? mA : 0.f;
        float m1 = (k1 < C) ? mA : 0.f;
        float e0 = expf(lrow[(k0 < C) ? k0 : 0]);
        float e1 = expf(lrow[(k1 < C) ? k1 : 0]);
        v2f a;
        a.x = e0 * m0;
        a.y = e1 * m1;
        c = __builtin_amdgcn_wmma_f32_16x16x4_f32(
                false, a, false, b, (short)0, c, false, false);
      }

      // D layout: lanes 0-15 hold M=r in VGPR r; lanes 16-31 hold M=r+8
      if (lane == 0) {
        #pragma unroll
        for (int r = 0; r < 8; ++r) {
          int row2 = wave * 16 + r;
          if (row2 < Mt) s_sumexp[row2] = c[r];
        }
      } else if (lane == 16) {
        #pragma unroll
        for (int r = 0; r < 8; ++r) {
          int row2 = wave * 16 + 8 + r;
          if (row2 < Mt) s_sumexp[row2] = c[r];
        }
      }
    }
  }
  __syncthreads();

  // ---- per-match losses ----
  double lsum = 0.0; int lacc = 0;
  if (tid < Mt) {
    const int t    = tid;
    const int pred = s_matchPred[t];
    const int lbl  = s_tlab[t];
    const float* lrow = confs + (size_t)pred * (size_t)C;

    float lse = logf(s_sumexp[t]);
    lsum += -(double)(lrow[lbl] - lse);        // cross-entropy term

    float bv = lrow[0]; int bi = 0;            // argmax (first occurrence on ties)
    for (int k = 1; k < C; ++k) { float x = lrow[k]; if (x > bv) { bv = x; bi = k; } }
    lacc = (bi == lbl);

    float4 pb = boxes[pred];
    float4 tb = s_tbox[t];
    float dsum = 0.f, d, ad;
    d = pb.x - tb.x; ad = fabsf(d); dsum += (ad < 1.f) ? 0.5f * d * d : ad - 0.5f;
    d = pb.y - tb.y; ad = fabsf(d); dsum += (ad < 1.f) ? 0.5f * d * d : ad - 0.5f;
    d = pb.z - tb.z; ad = fabsf(d); dsum += (ad < 1.f) ? 0.5f * d * d : ad - 0.5f;
    d = pb.w - tb.w; ad = fabsf(d); dsum += (ad < 1.f) ? 0.5f * d * d : ad - 0.5f;
    lsum += (double)(dsum * 0.25f);            // mean over 4 coords
  }

  rv[tid] = lsum; ri[tid] = lacc;
  __syncthreads();
  for (int off = TPB / 2; off > 0; off >>= 1) {
    if (tid < off) { rv[tid] += rv[tid + off]; ri[tid] += ri[tid + off]; }
    __syncthreads();
  }
  if (tid == 0) {
    out[0] = (float)rv[0];   // reg_loss + cls_loss
    out[1] = (float)ri[0];   // train_acc
  }
}

extern "C" void kernel_launch(void* const* d_in, const int* in_sizes, int n_in,
                              void* d_out, int out_size, void* d_ws, size_t ws_size,
                              hipStream_t stream) {
  const float4* boxes   = (const float4*)d_in[0];
  const float*  confs   = (const float*) d_in[1];
  const float4* tlocs   = (const float4*)d_in[2];
  const int*    tlabels = (const int*)   d_in[3];
  float* out = (float*)d_out;

  const int N  = in_sizes[0] / 4;      // 100000 predictions
  const int C  = in_sizes[1] / N;      // 81 classes
  const int Mt = in_sizes[3];          // 128 targets

  // carve workspace (~2.5 MB total)
  char* w = (char*)d_ws;
  size_t off = 0;
  auto carve = [&](size_t bytes) -> void* {
    void* ptr = w + off;
    off = (off + bytes + 255) & ~(size_t)255;
    return ptr;
  };
  double*        v_   = (double*)       carve((size_t)(N + 1) * sizeof(double));
  double*        s_   = (double*)       carve((size_t)(N + 1) * sizeof(double));
  int*           way_ = (int*)          carve((size_t)(N + 1) * sizeof(int));
  int*           p_   = (int*)          carve((size_t)(N + 1) * sizeof(int));
  unsigned char* u_   = (unsigned char*)carve((size_t)(N + 1));
  (void)ws_size; (void)n_in; (void)out_size;

  multibox_hungarian_loss<<<1, TPB, 0, stream>>>(
      boxes, confs, tlocs, tlabels, out, v_, s_, way_, p_, u_, N, C, Mt);
}